// ASAR_51659866636384
// MI455X (gfx1250) — compile-verified
//
#include <hip/hip_runtime.h>
#include <hip/hip_bf16.h>

typedef __attribute__((ext_vector_type(2))) float v2f;
typedef __attribute__((ext_vector_type(8))) float v8f;

#define NPTS 131072
#define KCLUST 2048
#define DIM 128
#define NITER 5
#define NEG_INF (-3.402823466e38f)

// ---------------------------------------------------------------------------
// halfnorm[k] = 0.5 * |C[k]|^2
// ---------------------------------------------------------------------------
__global__ void __launch_bounds__(256)
ASAR_halfnorm_kernel(const float* __restrict__ C, float* __restrict__ hn) {
    int k = blockIdx.x * blockDim.x + threadIdx.x;
    if (k >= KCLUST) return;
    const float* row = C + (size_t)k * DIM;
    float s = 0.0f;
#pragma unroll
    for (int d = 0; d < DIM; d += 4) {
        float4 v = *(const float4*)(row + d);
        s += v.x * v.x + v.y * v.y + v.z * v.z + v.w * v.w;
    }
    hn[k] = 0.5f * s;
}

// ---------------------------------------------------------------------------
// Assignment: one wave handles TWO 16-row tiles of X (register blocking).
// Each B fragment (centroid data) feeds 2 independent WMMAs. B fragments go
// through an explicit 8-slot rotating prefetch buffer so ~8 loads stay in
// flight while WMMAs execute (Round 3 collapsed to depth-1, wait-0 loads).
// Tracks per-row argmax of (x.c - 0.5|c|^2) == argmax of reference sim.
// ---------------------------------------------------------------------------
__global__ void __launch_bounds__(128, 1)
ASAR_assign_kernel(const float* __restrict__ X,
                   const float* __restrict__ C,
                   const float* __restrict__ hn,
                   int* __restrict__ labels) {
    const int lane = threadIdx.x & 31;
    const int wv   = threadIdx.x >> 5;
    const int row_base = (blockIdx.x * 4 + wv) * 32;   // two 16-row tiles
    const int rl   = lane & 15;                        // row (A) / col (B)
    const int half = lane >> 4;                        // 0 or 1
    const int kh   = half * 2;                         // k sub-offset (ISA layout)

    // Preload two A panels (64 VGPRs each)
    const float* xrow0 = X + (size_t)(row_base + rl) * DIM + kh;
    const float* xrow1 = X + (size_t)(row_base + 16 + rl) * DIM + kh;
    v2f a0[32], a1[32];
#pragma unroll
    for (int s = 0; s < 32; ++s) {
        a0[s] = *(const v2f*)(xrow0 + s * 4);
        a1[s] = *(const v2f*)(xrow1 + s * 4);
    }

    float best0[8], best1[8];
    int   bidx0[8], bidx1[8];
#pragma unroll
    for (int r = 0; r < 8; ++r) {
        best0[r] = NEG_INF; bidx0[r] = 0;
        best1[r] = NEG_INF; bidx1[r] = 0;
    }

    for (int ct = 0; ct < KCLUST / 16; ++ct) {
        const int col = ct * 16 + rl;                  // global cluster column
        const float* crow = C + (size_t)col * DIM + kh;
        const float bias = hn[col];

        // 8-deep rotating B prefetch buffer (fully static indexing).
        v2f bb[8];
#pragma unroll
        for (int p = 0; p < 8; ++p)
            bb[p] = *(const v2f*)(crow + p * 4);

        v8f acc0 = {}, acc1 = {};
#pragma unroll
        for (int s = 0; s < 32; ++s) {
            v2f b = bb[s & 7];
            if (s + 8 < 32)
                bb[s & 7] = *(const v2f*)(crow + (s + 8) * 4);
            acc0 = __builtin_amdgcn_wmma_f32_16x16x4_f32(
                false, a0[s], false, b, (short)0, acc0, false, false);
            acc1 = __builtin_amdgcn_wmma_f32_16x16x4_f32(
                false, a1[s], false, b, (short)0, acc1, false, false);
        }

        // acc[r] holds row (r + 8*half), column col. Strict '>' keeps the
        // first (lowest-index) max within a lane, matching jnp.argmax.
#pragma unroll
        for (int r = 0; r < 8; ++r) {
            float sc0 = acc0[r] - bias;
            if (sc0 > best0[r]) { best0[r] = sc0; bidx0[r] = col; }
            float sc1 = acc1[r] - bias;
            if (sc1 > best1[r]) { best1[r] = sc1; bidx1[r] = col; }
        }
    }

    // Cross-lane argmax over the 16 columns held by each half-group.
#pragma unroll
    for (int r = 0; r < 8; ++r) {
        float v = best0[r];
        int   i = bidx0[r];
#pragma unroll
        for (int m = 1; m < 16; m <<= 1) {
            float ov = __shfl_xor(v, m, 32);
            int   oi = __shfl_xor(i, m, 32);
            if (ov > v || (ov == v && oi < i)) { v = ov; i = oi; }
        }
        if (rl == 0)
            labels[row_base + r + 8 * half] = i;

        float v1 = best1[r];
        int   i1 = bidx1[r];
#pragma unroll
        for (int m = 1; m < 16; m <<= 1) {
            float ov = __shfl_xor(v1, m, 32);
            int   oi = __shfl_xor(i1, m, 32);
            if (ov > v1 || (ov == v1 && oi < i1)) { v1 = ov; i1 = oi; }
        }
        if (rl == 0)
            labels[row_base + 16 + r + 8 * half] = i1;
    }
}

// ---------------------------------------------------------------------------
// Accumulate: one wave per point; lane handles 4 dims (float4, coalesced).
// ---------------------------------------------------------------------------
__global__ void __launch_bounds__(256)
ASAR_accum_kernel(const float* __restrict__ X,
                  const int* __restrict__ labels,
                  float* __restrict__ sums,
                  float* __restrict__ counts) {
    const int lane = threadIdx.x & 31;
    const int wv   = threadIdx.x >> 5;
    const int pt   = blockIdx.x * 8 + wv;
    const int lbl  = labels[pt];
    const float4 v = *(const float4*)(X + (size_t)pt * DIM + lane * 4);
    float* dst = sums + (size_t)lbl * DIM + lane * 4;
    atomicAdd(dst + 0, v.x);
    atomicAdd(dst + 1, v.y);
    atomicAdd(dst + 2, v.z);
    atomicAdd(dst + 3, v.w);
    if (lane == 0) atomicAdd(counts + lbl, 1.0f);
}

// ---------------------------------------------------------------------------
// Update: C[k][d] = counts[k] > 0 ? sums[k][d]/max(counts[k],1) : 0
// ---------------------------------------------------------------------------
__global__ void __launch_bounds__(256)
ASAR_update_kernel(const float* __restrict__ sums,
                   const float* __restrict__ counts,
                   float* __restrict__ C) {
    int idx = blockIdx.x * blockDim.x + threadIdx.x;   // 0 .. K*DIM-1
    int k = idx >> 7;
    float c = counts[k];
    C[idx] = (c > 0.0f) ? sums[idx] / fmaxf(c, 1.0f) : 0.0f;
}

// ---------------------------------------------------------------------------
extern "C" void kernel_launch(void* const* d_in, const int* in_sizes, int n_in,
                              void* d_out, int out_size, void* d_ws, size_t ws_size,
                              hipStream_t stream) {
    const float* X    = (const float*)d_in[0];   // [131072, 128]
    const float* init = (const float*)d_in[1];   // [2048, 128]

    float* out_centroids = (float*)d_out;                        // 262144 f32
    int*   out_labels    = (int*)(out_centroids + KCLUST * DIM); // 131072 i32

    // Workspace layout
    float* C      = (float*)d_ws;                // 262144 f32
    float* sums   = C + KCLUST * DIM;            // 262144 f32
    float* counts = sums + KCLUST * DIM;         // 2048 f32 (contig w/ sums)
    float* hn     = counts + KCLUST;             // 2048 f32

    hipMemcpyAsync(C, init, (size_t)KCLUST * DIM * sizeof(float),
                   hipMemcpyDeviceToDevice, stream);

    for (int it = 0; it < NITER; ++it) {
        ASAR_halfnorm_kernel<<<(KCLUST + 255) / 256, 256, 0, stream>>>(C, hn);
        // 1024 blocks * 4 waves * 2 tiles * 16 rows = 131072 rows
        ASAR_assign_kernel<<<NPTS / 128, 128, 0, stream>>>(X, C, hn, out_labels);
        hipMemsetAsync(sums, 0, (size_t)(KCLUST * DIM + KCLUST) * sizeof(float),
                       stream);
        ASAR_accum_kernel<<<NPTS / 8, 256, 0, stream>>>(X, out_labels, sums, counts);
        ASAR_update_kernel<<<(KCLUST * DIM) / 256, 256, 0, stream>>>(sums, counts, C);
    }

    hipMemcpyAsync(out_centroids, C, (size_t)KCLUST * DIM * sizeof(float),
                   hipMemcpyDeviceToDevice, stream);
}